// GNet_45320494907545
// MI455X (gfx1250) — compile-verified
//
#include <hip/hip_runtime.h>
#include <hip/hip_bf16.h>
#include <stddef.h>

// Problem constants (match reference)
#define BB 512
#define LL 512
#define CC 64
#define CT 16
#define CO 8
#define IN_C 88     // CT + CO + CC
#define HH 128
#define G3 384      // 3*H
#define EPSI 1e-5f

typedef __attribute__((ext_vector_type(2))) float v2f;
typedef __attribute__((ext_vector_type(8))) float v8f;
typedef __attribute__((ext_vector_type(4))) int   v4i;

typedef __attribute__((address_space(1))) v4i* gbl_v4i_p;   // global int4*
typedef __attribute__((address_space(3))) v4i* lds_v4i_p;   // LDS int4*

// Async global->LDS path (CDNA5 GLOBAL_LOAD_ASYNC_TO_LDS_B128 + ASYNCcnt),
// guarded so the file still compiles if this toolchain lacks the builtins.
#if defined(__has_builtin)
#  if __has_builtin(__builtin_amdgcn_global_load_async_to_lds_b128) && \
      __has_builtin(__builtin_amdgcn_s_wait_asynccnt)
#    define HAVE_ASYNC_LDS 1
#  endif
#endif
#ifndef HAVE_ASYNC_LDS
#  define HAVE_ASYNC_LDS 0
#endif

// ---------------------------------------------------------------------------
// Kernel 1: time-norm over L axis for each (b, channel); writes concatenated
// normalized input X[(b*L + l)*88 + col], concat order [treat, outcome, cov].
// ---------------------------------------------------------------------------
__global__ void tnorm_kernel(const float* __restrict__ treat,
                             const float* __restrict__ outc,
                             const float* __restrict__ cov,
                             const float* __restrict__ g_t, const float* __restrict__ bt,
                             const float* __restrict__ g_o, const float* __restrict__ bo,
                             const float* __restrict__ g_c, const float* __restrict__ bc,
                             float* __restrict__ X) {
  int tid = blockIdx.x * blockDim.x + threadIdx.x;
  if (tid >= BB * IN_C) return;
  int b = tid / IN_C;
  int col = tid % IN_C;

  const float* src; int C; int ch; float g, be;
  if (col < CT)            { src = treat; C = CT; ch = col;            g = g_t[ch]; be = bt[ch]; }
  else if (col < CT + CO)  { src = outc;  C = CO; ch = col - CT;       g = g_o[ch]; be = bo[ch]; }
  else                     { src = cov;   C = CC; ch = col - CT - CO;  g = g_c[ch]; be = bc[ch]; }

  const float* p = src + (size_t)b * LL * C + ch;
  float s = 0.f, s2 = 0.f;
  for (int l = 0; l < LL; ++l) {
    float x = p[(size_t)l * C];
    s += x; s2 += x * x;
  }
  float mean = s * (1.f / LL);
  float var  = s2 * (1.f / LL) - mean * mean;
  float rstd = rsqrtf(var + EPSI);
  float scale = rstd * g;

  float* xo = X + (size_t)b * LL * IN_C + col;
  for (int l = 0; l < LL; ++l) {
    float x = p[(size_t)l * C];
    xo[(size_t)l * IN_C] = (x - mean) * scale + be;
  }
}

// ---------------------------------------------------------------------------
// Kernel 2: G = X @ W_ih^T + b_ih, X:(B*L, 88), W_ih:(384, 88), G:(B*L, 384).
// One wave per 16x16 output tile; K=88 in 22 steps of V_WMMA_F32_16X16X4_F32.
// ---------------------------------------------------------------------------
__global__ void gemm_ih_kernel(const float* __restrict__ X,
                               const float* __restrict__ Wih,
                               const float* __restrict__ b_ih,
                               float* __restrict__ G) {
  int wave = threadIdx.x >> 5;
  int lane = threadIdx.x & 31;
  int tile = blockIdx.x * 8 + wave;          // 393216 tiles total
  int mt = tile / 24;                        // 16384 M tiles
  int nt = tile % 24;                        // 24 N tiles (384/16)

  int n16 = lane & 15;
  int ks  = (lane >> 4) << 1;                // 0 or 2

  const float* ap = X   + (size_t)(mt * 16 + n16) * IN_C + ks;
  const float* bp = Wih + (size_t)(nt * 16 + n16) * IN_C + ks;

  v8f acc = {};
#pragma unroll
  for (int kk = 0; kk < 22; ++kk) {
    v2f a = *(const v2f*)(ap + kk * 4);
    v2f b = *(const v2f*)(bp + kk * 4);
    acc = __builtin_amdgcn_wmma_f32_16x16x4_f32(false, a, false, b,
                                                (short)0, acc, false, false);
  }

  int mrow = mt * 16 + ((lane >> 4) << 3);   // +0 or +8
  int ncol = nt * 16 + n16;
  float bias = b_ih[ncol];
#pragma unroll
  for (int i = 0; i < 8; ++i)
    G[(size_t)(mrow + i) * G3 + ncol] = acc[i] + bias;
}

// ---------------------------------------------------------------------------
// Kernel 3: sequential GRU scan. 32 blocks x 768 threads (24 waves).
// Each block owns 16 batch rows; h lives in LDS; W_hh B-frags live in VGPRs
// (wave w owns gate columns [16w, 16w+16), 32 k-steps x v2f = 64 VGPRs).
// Per step: 32 fp32 WMMAs/wave from LDS h; next step's G rows are staged into
// LDS with async global->LDS copies (double buffered) while WMMAs run.
// ---------------------------------------------------------------------------
__global__ void __launch_bounds__(768, 1)
gru_scan_kernel(const float* __restrict__ G,
                const float* __restrict__ Whh,
                const float* __restrict__ b_hh,
                float* __restrict__ out_o,
                float* __restrict__ out_c) {
  __shared__ float sh_h[16 * HH];         // 8 KB
  __shared__ float sh_gh[16 * G3];        // 24 KB
  __shared__ float sh_bhh[G3];            // 1.5 KB
#if HAVE_ASYNC_LDS
  __shared__ __align__(16) float sh_gi[2][16 * G3];  // 2 x 24 KB double buffer
#endif

  int tid  = threadIdx.x;
  int wave = tid >> 5;
  int lane = tid & 31;
  int mBase = blockIdx.x * 16;

  int n16 = lane & 15;
  int ks  = (lane >> 4) << 1;             // 0 or 2
  int ncol0 = wave * 16;

  // Preload this wave's W_hh slice as 32 B-fragments (held in VGPRs).
  v2f bfrag[32];
  const float* wp = Whh + (size_t)(ncol0 + n16) * HH + ks;
#pragma unroll
  for (int kk = 0; kk < 32; ++kk)
    bfrag[kk] = *(const v2f*)(wp + kk * 4);

  // h = 0; b_hh -> LDS
  for (int i = tid; i < 16 * HH; i += 768) sh_h[i] = 0.f;
  for (int i = tid; i < G3; i += 768) sh_bhh[i] = b_hh[i];

#if HAVE_ASYNC_LDS
  // Stage G rows for timestep t into buffer buf: 16 rows x 1536 B = 1536 B128s.
  auto stage = [&](int t, int buf) {
#pragma unroll
    for (int u = 0; u < 2; ++u) {
      int e = tid + u * 768;              // 0..1535
      int m = e / 96;                     // 96 x 16B per row
      int q = e % 96;
      const float* src = G + ((size_t)(mBase + m) * LL + t) * G3 + q * 4;
      float* dst = &sh_gi[buf][m * G3 + q * 4];
      __builtin_amdgcn_global_load_async_to_lds_b128(
          (gbl_v4i_p)(void*)src, (lds_v4i_p)(void*)dst,
          /*offset=*/0, /*cpol=*/0);
    }
  };
  stage(0, 0);
#endif
  __syncthreads();

  int mrow = (lane >> 4) << 3;            // C/D frag row base: 0 or 8

  for (int t = 0; t < LL; ++t) {
    // gh tile = h @ W_hh^T  (this wave's 16 columns)
    v8f acc = {};
#pragma unroll
    for (int kk = 0; kk < 32; ++kk) {
      v2f a = *(const v2f*)&sh_h[n16 * HH + kk * 4 + ks];
      acc = __builtin_amdgcn_wmma_f32_16x16x4_f32(false, a, false, bfrag[kk],
                                                  (short)0, acc, false, false);
    }
#pragma unroll
    for (int i = 0; i < 8; ++i)
      sh_gh[(mrow + i) * G3 + ncol0 + n16] = acc[i];

#if HAVE_ASYNC_LDS
    // Kick next step's staging, then ensure current buffer has landed.
    if (t + 1 < LL) {
      stage(t + 1, (t + 1) & 1);
      __builtin_amdgcn_s_wait_asynccnt(2);   // <=2 outstanding: stage(t) done
    } else {
      __builtin_amdgcn_s_wait_asynccnt(0);
    }
#else
    // Fallback: prefetch next step's G rows (16 rows x 1536 B).
    if (t + 1 < LL) {
      for (int pf = tid; pf < 16 * 12; pf += 768) {
        int m = pf / 12, cl = pf % 12;
        const char* a =
            (const char*)(G + ((size_t)(mBase + m) * LL + (t + 1)) * G3) + cl * 128;
        __builtin_prefetch(a, 0, 1);
      }
    }
#endif
    __syncthreads();

    // Fused gates + h update + output split.
#if HAVE_ASYNC_LDS
    const float* gi = sh_gi[t & 1];
#endif
    for (int idx = tid; idx < 16 * HH; idx += 768) {
      int m = idx >> 7;
      int j = idx & 127;
#if HAVE_ASYNC_LDS
      float i_r = gi[m * G3 + j];
      float i_z = gi[m * G3 + HH + j];
      float i_n = gi[m * G3 + 2 * HH + j];
#else
      size_t grow = ((size_t)(mBase + m) * LL + t) * G3;
      float i_r = G[grow + j];
      float i_z = G[grow + HH + j];
      float i_n = G[grow + 2 * HH + j];
#endif
      float h_r = sh_gh[m * G3 + j]          + sh_bhh[j];
      float h_z = sh_gh[m * G3 + HH + j]     + sh_bhh[HH + j];
      float h_n = sh_gh[m * G3 + 2 * HH + j] + sh_bhh[2 * HH + j];
      float r  = 1.f / (1.f + __expf(-(i_r + h_r)));
      float z  = 1.f / (1.f + __expf(-(i_z + h_z)));
      float nn = tanhf(i_n + r * h_n);
      float hp = sh_h[m * HH + j];
      float hv = (1.f - z) * nn + z * hp;
      sh_h[m * HH + j] = hv;                 // own element only; barrier below
      size_t orow = (size_t)(mBase + m) * LL + t;
      if (j < CO) out_o[orow * CO + j] = hv;
      else        out_c[orow * (HH - CO) + (j - CO)] = hv;
    }
    __syncthreads();
  }
}

// ---------------------------------------------------------------------------
// Host launcher
// ---------------------------------------------------------------------------
extern "C" void kernel_launch(void* const* d_in, const int* in_sizes, int n_in,
                              void* d_out, int out_size, void* d_ws, size_t ws_size,
                              hipStream_t stream) {
  (void)in_sizes; (void)n_in; (void)out_size; (void)ws_size;

  const float* cov   = (const float*)d_in[0];   // (B,L,64)
  const float* treat = (const float*)d_in[1];   // (B,L,16)
  const float* outc  = (const float*)d_in[2];   // (B,L,8)
  const float* W_ih  = (const float*)d_in[3];   // (384,88)
  const float* W_hh  = (const float*)d_in[4];   // (384,128)
  const float* b_ih  = (const float*)d_in[5];   // (384,)
  const float* b_hh  = (const float*)d_in[6];   // (384,)
  const float* g_t   = (const float*)d_in[7];
  const float* bt    = (const float*)d_in[8];
  const float* g_o   = (const float*)d_in[9];
  const float* bo    = (const float*)d_in[10];
  const float* g_c   = (const float*)d_in[11];
  const float* bc    = (const float*)d_in[12];

  float* X = (float*)d_ws;                                   // (B*L, 88)
  float* G = X + (size_t)BB * LL * IN_C;                     // (B*L, 384)

  float* out_o = (float*)d_out;                              // (B,L,8)
  float* out_c = out_o + (size_t)BB * LL * CO;               // (B,L,120)

  // 1) time-norm + concat:  512*88 threads
  tnorm_kernel<<<(BB * IN_C) / 256, 256, 0, stream>>>(
      treat, outc, cov, g_t, bt, g_o, bo, g_c, bc, X);

  // 2) input projection GEMM: 16384*24 tiles, 8 waves/block
  gemm_ih_kernel<<<(16384 * 24) / 8, 256, 0, stream>>>(X, W_ih, b_ih, G);

  // 3) sequential scan: 32 blocks of 24 waves
  gru_scan_kernel<<<BB / 16, 768, 0, stream>>>(G, W_hh, b_hh, out_o, out_c);
}